// RetrieverReachabilityLoss_14482629722496
// MI455X (gfx1250) — compile-verified
//
#include <hip/hip_runtime.h>
#include <hip/hip_bf16.h>

// ---------------------------------------------------------------------------
// Segmented softmax listwise loss + segmented-mean BCE  (E=16.7M, G=8192)
//
// Roofline: pure streaming reduction, ~1 flop/byte -> HBM-bound.
//   Pass A: read logits(67MB) + int64 seg(134MB), emit per-graph max
//           + compacted u16 seg ids (33MB write, L2-resident).
//   Pass B: read logits+targets(134MB) + seg16(33MB, mostly L2 hit).
//   Total HBM ~370-400MB  ->  ~16-17us at 23.3 TB/s.
// Sorted segment ids -> per-thread run accumulation, ~1 atomic per run
// instead of per edge (~5M atomics total vs 84M naive).
// Finalize: wave32 sum via V_WMMA_F32_16X16X4_F32 ones-trick (exact f32).
// ---------------------------------------------------------------------------

typedef float    v2f   __attribute__((ext_vector_type(2)));
typedef float    v8f   __attribute__((ext_vector_type(8)));
typedef float    f32x4 __attribute__((ext_vector_type(4)));
typedef int      i32x4 __attribute__((ext_vector_type(4)));
typedef unsigned u32x4 __attribute__((ext_vector_type(4)));

#define KPT 16          // edges per thread in streaming passes
#define BLK 256         // 8 wave32 waves

// Monotonic float <-> ordered-uint encoding for atomicMax on floats.
__device__ __forceinline__ unsigned f2ord(float f) {
    int b = __float_as_int(f);
    return (unsigned)(b ^ ((b >> 31) | 0x80000000));
}
__device__ __forceinline__ float ord2f(unsigned u) {
    int b = (u & 0x80000000u) ? (int)(u ^ 0x80000000u) : (int)(~u);
    return __int_as_float(b);
}
#define ORD_NEG_INF 0x007FFFFFu   // f2ord(-inf)

// ---------------------------------------------------------------------------
// Wave32 all-lane sum via V_WMMA_F32_16X16X4_F32 with B = ones.
// D[m][n] = rowsum(A); lane sums its 8 D regs -> lanes 0-15 = rows 0-7,
// lanes 16-31 = rows 8-15; shfl_xor(16) combines halves. EXEC must be ~0.
// ---------------------------------------------------------------------------
__device__ __forceinline__ float wave_sum_wmma(float x) {
    v2f a; a.x = x;    a.y = 0.0f;
    v2f b; b.x = 1.0f; b.y = 1.0f;
    v8f c = {};
    c = __builtin_amdgcn_wmma_f32_16x16x4_f32(
            false, a, false, b, (short)0, c, false, false);
    float s = c[0] + c[1] + c[2] + c[3] + c[4] + c[5] + c[6] + c[7];
    s += __shfl_xor(s, 16, 32);
    return s;
}

// ---------------------------------------------------------------------------
__global__ void k_init(unsigned* __restrict__ umax, float* __restrict__ zeros,
                       int G, int ztotal) {
    int i = blockIdx.x * blockDim.x + threadIdx.x;
    if (i < G) umax[i] = ORD_NEG_INF;
    if (i < ztotal) zeros[i] = 0.0f;
}

// ---------------------------------------------------------------------------
// Pass A: per-graph max of logits (TEMPERATURE==1 -> scaled==logits).
// Optionally compacts int64 seg ids to u16 for pass B.
// NT loads: this pass's stream must not evict the seg16 output from L2.
// ---------------------------------------------------------------------------
template <bool COMPACT>
__global__ void k_pass_a(const float* __restrict__ logits,
                         const int* __restrict__ seg2,          // int64 low word at 2*i
                         unsigned* __restrict__ umax,
                         unsigned short* __restrict__ seg16,
                         long long E) {
    long long base = ((long long)blockIdx.x * blockDim.x + threadIdx.x) * (long long)KPT;
    if (base >= E) return;

    int cur = -1;
    float m = 0.0f;

    if (base + KPT <= E) {
        const f32x4* l4 = (const f32x4*)(logits + base);
        const i32x4* s4 = (const i32x4*)(seg2 + 2 * base);
        u32x4*       p4 = COMPACT ? (u32x4*)(seg16 + base) : nullptr;
#pragma unroll
        for (int c = 0; c < KPT / 8; ++c) {
            f32x4 L0 = __builtin_nontemporal_load(l4 + 2 * c);
            f32x4 L1 = __builtin_nontemporal_load(l4 + 2 * c + 1);
            i32x4 sA = __builtin_nontemporal_load(s4 + 4 * c);
            i32x4 sB = __builtin_nontemporal_load(s4 + 4 * c + 1);
            i32x4 sC = __builtin_nontemporal_load(s4 + 4 * c + 2);
            i32x4 sD = __builtin_nontemporal_load(s4 + 4 * c + 3);
            int   ss[8] = {sA.x, sA.z, sB.x, sB.z, sC.x, sC.z, sD.x, sD.z};
            float ll[8] = {L0.x, L0.y, L0.z, L0.w, L1.x, L1.y, L1.z, L1.w};
            if (COMPACT) {
                u32x4 P;
                P.x = (unsigned)(ss[0] & 0xFFFF) | ((unsigned)(ss[1] & 0xFFFF) << 16);
                P.y = (unsigned)(ss[2] & 0xFFFF) | ((unsigned)(ss[3] & 0xFFFF) << 16);
                P.z = (unsigned)(ss[4] & 0xFFFF) | ((unsigned)(ss[5] & 0xFFFF) << 16);
                P.w = (unsigned)(ss[6] & 0xFFFF) | ((unsigned)(ss[7] & 0xFFFF) << 16);
                p4[c] = P;   // RT store: keep resident in L2 for pass B
            }
#pragma unroll
            for (int j = 0; j < 8; ++j) {
                if (ss[j] != cur) {
                    if (cur >= 0) atomicMax(&umax[cur], f2ord(m));
                    cur = ss[j];
                    m = ll[j];
                } else {
                    m = fmaxf(m, ll[j]);
                }
            }
        }
    } else {
        for (long long i = base; i < E; ++i) {
            int s = seg2[2 * i];
            float l = logits[i];
            if (COMPACT) seg16[i] = (unsigned short)s;
            if (s != cur) {
                if (cur >= 0) atomicMax(&umax[cur], f2ord(m));
                cur = s;
                m = l;
            } else {
                m = fmaxf(m, l);
            }
        }
    }
    if (cur >= 0) atomicMax(&umax[cur], f2ord(m));
}

// ---------------------------------------------------------------------------
// Pass B helpers
// ---------------------------------------------------------------------------
__device__ __forceinline__ void edge_accum(float l, float t, float mb,
                                           float& ad, float& an, float& ab, float& ac) {
    float w = fminf(fmaxf(t, 0.0f), 1.0f);
    float e = __expf(l - mb);
    ad += e;
    an += e * w;
    ac += 1.0f;
    // softplus(-|l|) via hw exp/log; abs err <= 2e-9 vs log1pf where 1+y rounds to 1
    ab += fmaxf(l, 0.0f) - l * t + __logf(1.0f + __expf(-fabsf(l)));
}

__device__ __forceinline__ void flush_sums(int cur, float ad, float an, float ab, float ac,
                                           float* __restrict__ denom, float* __restrict__ numer,
                                           float* __restrict__ bces,  float* __restrict__ cnts) {
    if (cur >= 0) {
        atomicAdd(&denom[cur], ad);
        atomicAdd(&numer[cur], an);
        atomicAdd(&bces[cur],  ab);
        atomicAdd(&cnts[cur],  ac);
    }
}

// Pass B (compact): seg ids from u16 array (L2-resident); NT float streams.
__global__ void k_pass_b16(const float* __restrict__ logits,
                           const float* __restrict__ targets,
                           const unsigned short* __restrict__ seg16,
                           const unsigned* __restrict__ umax,
                           float* __restrict__ denom, float* __restrict__ numer,
                           float* __restrict__ bces,  float* __restrict__ cnts,
                           long long E) {
    long long base = ((long long)blockIdx.x * blockDim.x + threadIdx.x) * (long long)KPT;
    if (base >= E) return;

    int cur = -1;
    float mb = 0.0f;
    float ad = 0.0f, an = 0.0f, ab = 0.0f, ac = 0.0f;

    if (base + KPT <= E) {
        const f32x4* l4 = (const f32x4*)(logits + base);
        const f32x4* t4 = (const f32x4*)(targets + base);
        const u32x4* sv = (const u32x4*)(seg16 + base);
#pragma unroll
        for (int c = 0; c < KPT / 8; ++c) {
            u32x4 S  = sv[c];                                  // RT: want the L2 hit
            f32x4 L0 = __builtin_nontemporal_load(l4 + 2 * c);
            f32x4 L1 = __builtin_nontemporal_load(l4 + 2 * c + 1);
            f32x4 T0 = __builtin_nontemporal_load(t4 + 2 * c);
            f32x4 T1 = __builtin_nontemporal_load(t4 + 2 * c + 1);
            int ss[8] = {(int)(S.x & 0xFFFFu), (int)(S.x >> 16),
                         (int)(S.y & 0xFFFFu), (int)(S.y >> 16),
                         (int)(S.z & 0xFFFFu), (int)(S.z >> 16),
                         (int)(S.w & 0xFFFFu), (int)(S.w >> 16)};
            float ll[8] = {L0.x, L0.y, L0.z, L0.w, L1.x, L1.y, L1.z, L1.w};
            float tt[8] = {T0.x, T0.y, T0.z, T0.w, T1.x, T1.y, T1.z, T1.w};
#pragma unroll
            for (int j = 0; j < 8; ++j) {
                if (ss[j] != cur) {
                    flush_sums(cur, ad, an, ab, ac, denom, numer, bces, cnts);
                    cur = ss[j];
                    mb = ord2f(umax[cur]);
                    ad = an = ab = ac = 0.0f;
                }
                edge_accum(ll[j], tt[j], mb, ad, an, ab, ac);
            }
        }
    } else {
        for (long long i = base; i < E; ++i) {
            int s = (int)seg16[i];
            if (s != cur) {
                flush_sums(cur, ad, an, ab, ac, denom, numer, bces, cnts);
                cur = s;
                mb = ord2f(umax[cur]);
                ad = an = ab = ac = 0.0f;
            }
            edge_accum(logits[i], targets[i], mb, ad, an, ab, ac);
        }
    }
    flush_sums(cur, ad, an, ab, ac, denom, numer, bces, cnts);
}

// Pass B (fallback, ws too small for seg16): read int64 seg ids directly.
__global__ void k_pass_b64(const float* __restrict__ logits,
                           const float* __restrict__ targets,
                           const int* __restrict__ seg2,
                           const unsigned* __restrict__ umax,
                           float* __restrict__ denom, float* __restrict__ numer,
                           float* __restrict__ bces,  float* __restrict__ cnts,
                           long long E) {
    long long base = ((long long)blockIdx.x * blockDim.x + threadIdx.x) * (long long)KPT;
    if (base >= E) return;

    int cur = -1;
    float mb = 0.0f;
    float ad = 0.0f, an = 0.0f, ab = 0.0f, ac = 0.0f;

    if (base + KPT <= E) {
        const f32x4* l4 = (const f32x4*)(logits + base);
        const f32x4* t4 = (const f32x4*)(targets + base);
        const i32x4* s4 = (const i32x4*)(seg2 + 2 * base);
#pragma unroll
        for (int c = 0; c < KPT / 8; ++c) {
            f32x4 L0 = __builtin_nontemporal_load(l4 + 2 * c);
            f32x4 L1 = __builtin_nontemporal_load(l4 + 2 * c + 1);
            f32x4 T0 = __builtin_nontemporal_load(t4 + 2 * c);
            f32x4 T1 = __builtin_nontemporal_load(t4 + 2 * c + 1);
            i32x4 sA = __builtin_nontemporal_load(s4 + 4 * c);
            i32x4 sB = __builtin_nontemporal_load(s4 + 4 * c + 1);
            i32x4 sC = __builtin_nontemporal_load(s4 + 4 * c + 2);
            i32x4 sD = __builtin_nontemporal_load(s4 + 4 * c + 3);
            int   ss[8] = {sA.x, sA.z, sB.x, sB.z, sC.x, sC.z, sD.x, sD.z};
            float ll[8] = {L0.x, L0.y, L0.z, L0.w, L1.x, L1.y, L1.z, L1.w};
            float tt[8] = {T0.x, T0.y, T0.z, T0.w, T1.x, T1.y, T1.z, T1.w};
#pragma unroll
            for (int j = 0; j < 8; ++j) {
                if (ss[j] != cur) {
                    flush_sums(cur, ad, an, ab, ac, denom, numer, bces, cnts);
                    cur = ss[j];
                    mb = ord2f(umax[cur]);
                    ad = an = ab = ac = 0.0f;
                }
                edge_accum(ll[j], tt[j], mb, ad, an, ab, ac);
            }
        }
    } else {
        for (long long i = base; i < E; ++i) {
            int s = seg2[2 * i];
            if (s != cur) {
                flush_sums(cur, ad, an, ab, ac, denom, numer, bces, cnts);
                cur = s;
                mb = ord2f(umax[cur]);
                ad = an = ab = ac = 0.0f;
            }
            edge_accum(logits[i], targets[i], mb, ad, an, ab, ac);
        }
    }
    flush_sums(cur, ad, an, ab, ac, denom, numer, bces, cnts);
}

// ---------------------------------------------------------------------------
// Per-graph finalize + cross-graph reduction (max_per cancels in the log diff).
// ---------------------------------------------------------------------------
__global__ void k_graph(const float* __restrict__ denom, const float* __restrict__ numer,
                        const float* __restrict__ bces,  const float* __restrict__ cnts,
                        float* __restrict__ accum, int G) {
    int g = blockIdx.x * blockDim.x + threadIdx.x;
    float lw = 0.0f, np = 0.0f, bm = 0.0f;
    if (g < G) {
        float d = denom[g];
        float n = numer[g];
        if (n > 0.0f) {
            const float tiny = 1.17549435e-38f;  // finfo(f32).tiny
            lw = logf(fmaxf(d, tiny)) - logf(fmaxf(n, tiny));
            np = 1.0f;
        }
        bm = bces[g] / fmaxf(cnts[g], 1.0f);
    }
    // Divergence reconverged; EXEC all-ones -> WMMA legal.
    float s0 = wave_sum_wmma(lw);
    float s1 = wave_sum_wmma(np);
    float s2 = wave_sum_wmma(bm);
    if ((threadIdx.x & 31) == 0) {
        atomicAdd(&accum[0], s0);
        atomicAdd(&accum[1], s1);
        atomicAdd(&accum[2], s2);
    }
}

__global__ void k_final(const float* __restrict__ accum, float* __restrict__ out, float G) {
    if (threadIdx.x == 0 && blockIdx.x == 0) {
        float lw_sum = accum[0];
        float n_pos  = accum[1];
        float bce    = accum[2] / G;
        float listwise = (n_pos > 0.0f) ? (lw_sum / fmaxf(n_pos, 1.0f)) : 0.0f;
        out[0] = 1.0f * listwise + 0.5f * bce;   // LISTWISE_WEIGHT=1, BCE_WEIGHT=0.5
    }
}

// ---------------------------------------------------------------------------
extern "C" void kernel_launch(void* const* d_in, const int* in_sizes, int n_in,
                              void* d_out, int out_size, void* d_ws, size_t ws_size,
                              hipStream_t stream) {
    const float* logits  = (const float*)d_in[0];
    const float* targets = (const float*)d_in[1];
    const int*   seg2    = (const int*)d_in[2];  // int64 ids; low dword at 2*i
    (void)n_in; (void)out_size;

    const long long E = (long long)in_sizes[0];
    const int G = 8192;  // reference num_graphs (python scalar)

    // Workspace layout
    char* ws = (char*)d_ws;
    unsigned* umax  = (unsigned*)ws;                 // [G]
    float*    denom = (float*)(ws + 4ull * G);       // [G]
    float*    numer = denom + G;                     // [G]
    float*    bces  = numer + G;                     // [G]
    float*    cnts  = bces + G;                      // [G]
    float*    accum = cnts + G;                      // [4]
    const size_t small_bytes = 20ull * G + 16ull;    // 163,856 B (16B aligned)
    unsigned short* seg16 = (unsigned short*)(ws + small_bytes);
    const bool compact = ws_size >= small_bytes + 2ull * (size_t)E;
    const int ztotal = 4 * G + 4;

    float* out = (float*)d_out;

    long long threads = (E + KPT - 1) / KPT;
    int blocks = (int)((threads + BLK - 1) / BLK);

    // 1) init small workspace (harness does not re-zero between replays;
    //    seg16 is fully rewritten by pass A so it needs no init)
    {
        int n = (G > ztotal) ? G : ztotal;
        k_init<<<(n + BLK - 1) / BLK, BLK, 0, stream>>>(umax, denom, G, ztotal);
    }
    // 2) per-graph max (+ seg id compaction)
    if (compact)
        k_pass_a<true ><<<blocks, BLK, 0, stream>>>(logits, seg2, umax, seg16, E);
    else
        k_pass_a<false><<<blocks, BLK, 0, stream>>>(logits, seg2, umax, seg16, E);
    // 3) per-graph sums (softmax denom/numer, BCE sum, counts)
    if (compact)
        k_pass_b16<<<blocks, BLK, 0, stream>>>(logits, targets, seg16, umax,
                                               denom, numer, bces, cnts, E);
    else
        k_pass_b64<<<blocks, BLK, 0, stream>>>(logits, targets, seg2, umax,
                                               denom, numer, bces, cnts, E);
    // 4) per-graph finalize + WMMA wave reduction
    k_graph<<<(G + BLK - 1) / BLK, BLK, 0, stream>>>(denom, numer, bces, cnts, accum, G);
    // 5) scalar combine
    k_final<<<1, 32, 0, stream>>>(accum, out, (float)G);
}